// stage2_gcn_encoder_3298534883878
// MI455X (gfx1250) — compile-verified
//
#include <hip/hip_runtime.h>
#include <hip/hip_bf16.h>
#include <math.h>

#define N_NODES  50000
#define N_EDGES  800000
#define IN_F     128
#define HID_F    64
#define OUT_F    32
#define N_GRAPHS 128

#define ROW_TILES ((N_NODES + 15) / 16)   // 3125

typedef __attribute__((ext_vector_type(2))) float v2f;
typedef __attribute__((ext_vector_type(4))) float v4f;
typedef __attribute__((ext_vector_type(8))) float v8f;

// ---------------------------------------------------------------------------
// 1) Edge-degree accumulation: deg[dst] += ew  (self-loop +1 folded later)
// ---------------------------------------------------------------------------
__global__ void degree_kernel(const int* __restrict__ dst,
                              const float* __restrict__ ew,
                              float* __restrict__ deg) {
    int e = blockIdx.x * blockDim.x + threadIdx.x;
    if (e < N_EDGES) {
        atomicAdd(&deg[dst[e]], ew[e]);
    }
}

// ---------------------------------------------------------------------------
// 2) dinv = rsqrt(deg + 1)   (deg includes self-loop weight 1 => always > 0)
// ---------------------------------------------------------------------------
__global__ void dinv_kernel(const float* __restrict__ deg,
                            float* __restrict__ dinv) {
    int i = blockIdx.x * blockDim.x + threadIdx.x;
    if (i < N_NODES) {
        dinv[i] = rsqrtf(deg[i] + 1.0f);
    }
}

// ---------------------------------------------------------------------------
// 3) h = x @ W via V_WMMA_F32_16X16X4_F32.
//    One wave per 16-row strip; computes all 4 column tiles (16x64 output)
//    so each A fragment feeds 4 WMMAs (x traffic read exactly once).
//    Block = 256 threads = 8 waves = 8 row strips.
//    A layout (16x4 f32): lanes 0-15 -> K={k0,k0+1}, lanes 16-31 -> {k0+2,k0+3}
//    B layout (4x16 f32): mirrors A with N striped across lanes.
//    C/D: VGPR v -> M = v + 8*(lane>=16), N = lane&15.
// ---------------------------------------------------------------------------
__global__ void gemm1_wmma(const float* __restrict__ x,
                           const float* __restrict__ W,
                           float* __restrict__ h) {
    const int wave = threadIdx.x >> 5;
    const int lane = threadIdx.x & 31;
    const int rt   = blockIdx.x * 8 + wave;       // row tile
    if (rt >= ROW_TILES) return;                  // wave-uniform exit

    const int half = lane >> 4;                   // 0: lanes 0-15, 1: lanes 16-31
    const int l15  = lane & 15;
    const int row  = rt * 16 + l15;               // A-matrix row (M)

    v8f acc0 = {}, acc1 = {}, acc2 = {}, acc3 = {};
    const float* __restrict__ xrow = x + (size_t)row * IN_F;

    #pragma unroll 2
    for (int k0 = 0; k0 < IN_F; k0 += 4) {
        const int k = k0 + 2 * half;              // even -> 8B-aligned float2 load
        v2f a = *(const v2f*)(xrow + k);
        const float* __restrict__ w0 = W + (size_t)k * HID_F + l15;
        const float* __restrict__ w1 = w0 + HID_F;

        v2f b0; b0.x = w0[0];  b0.y = w1[0];
        v2f b1; b1.x = w0[16]; b1.y = w1[16];
        v2f b2; b2.x = w0[32]; b2.y = w1[32];
        v2f b3; b3.x = w0[48]; b3.y = w1[48];

        acc0 = __builtin_amdgcn_wmma_f32_16x16x4_f32(false, a, false, b0, (short)0, acc0, false, false);
        acc1 = __builtin_amdgcn_wmma_f32_16x16x4_f32(false, a, false, b1, (short)0, acc1, false, false);
        acc2 = __builtin_amdgcn_wmma_f32_16x16x4_f32(false, a, false, b2, (short)0, acc2, false, false);
        acc3 = __builtin_amdgcn_wmma_f32_16x16x4_f32(false, a, false, b3, (short)0, acc3, false, false);
    }

    #pragma unroll
    for (int v = 0; v < 8; ++v) {
        const size_t m = (size_t)(rt * 16 + v + 8 * half);
        float* __restrict__ hr = h + m * HID_F + l15;
        hr[0]  = acc0[v];
        hr[16] = acc1[v];
        hr[32] = acc2[v];
        hr[48] = acc3[v];
    }
}

// ---------------------------------------------------------------------------
// 4) Scatter messages: agg[dst] += h[src] * (dinv[src]*ew*dinv[dst])
//    16 threads per edge, float4 per thread: amortizes the scalar
//    src/dst/ew/dinv lookups 4x vs. one-feature-per-thread.
// ---------------------------------------------------------------------------
__global__ void scatter_kernel(const float* __restrict__ h,
                               const float* __restrict__ dinv,
                               const float* __restrict__ ew,
                               const int* __restrict__ src,
                               const int* __restrict__ dst,
                               float* __restrict__ agg) {
    size_t gid = (size_t)blockIdx.x * blockDim.x + threadIdx.x;
    if (gid >= (size_t)N_EDGES * (HID_F / 4)) return;
    const int e  = (int)(gid >> 4);
    const int f4 = (int)(gid & 15) * 4;
    const int s = src[e];
    const int d = dst[e];
    const float norm = dinv[s] * ew[e] * dinv[d];
    const v4f hv = *(const v4f*)(h + (size_t)s * HID_F + f4);
    float* __restrict__ ap = agg + (size_t)d * HID_F + f4;
    atomicAdd(ap + 0, hv.x * norm);
    atomicAdd(ap + 1, hv.y * norm);
    atomicAdd(ap + 2, hv.z * norm);
    atomicAdd(ap + 3, hv.w * norm);
}

// ---------------------------------------------------------------------------
// 5) Finalize nodes: add self-loop message + bias, tanh, write node_emb,
//    accumulate graph pooling sums/counts. float4 per thread.
// ---------------------------------------------------------------------------
__global__ void finalize_kernel(const float* __restrict__ h,
                                const float* __restrict__ dinv,
                                const float* __restrict__ b,
                                const int* __restrict__ batch,
                                float* __restrict__ node_out,   // also agg buffer
                                float* __restrict__ sums,
                                float* __restrict__ cnts) {
    size_t gid = (size_t)blockIdx.x * blockDim.x + threadIdx.x;
    if (gid >= (size_t)N_NODES * (HID_F / 4)) return;
    const int i  = (int)(gid >> 4);
    const int f4 = (int)(gid & 15) * 4;
    const float di  = dinv[i];
    const float di2 = di * di;

    const size_t off = (size_t)i * HID_F + f4;
    const v4f av = *(const v4f*)(node_out + off);
    const v4f hv = *(const v4f*)(h + off);
    const v4f bv = *(const v4f*)(b + f4);

    v4f t;
    t.x = tanhf(av.x + hv.x * di2 + bv.x);
    t.y = tanhf(av.y + hv.y * di2 + bv.y);
    t.z = tanhf(av.z + hv.z * di2 + bv.z);
    t.w = tanhf(av.w + hv.w * di2 + bv.w);
    *(v4f*)(node_out + off) = t;

    const int g = batch[i];
    float* __restrict__ sp = sums + (size_t)g * HID_F + f4;
    atomicAdd(sp + 0, t.x);
    atomicAdd(sp + 1, t.y);
    atomicAdd(sp + 2, t.z);
    atomicAdd(sp + 3, t.w);
    if (f4 == 0) atomicAdd(&cnts[g], 1.0f);
}

// ---------------------------------------------------------------------------
// 6) Mean pooling: ge = sums / max(cnts, 1)
// ---------------------------------------------------------------------------
__global__ void pool_mean_kernel(const float* __restrict__ sums,
                                 const float* __restrict__ cnts,
                                 float* __restrict__ ge) {
    int idx = blockIdx.x * blockDim.x + threadIdx.x;
    if (idx < N_GRAPHS * HID_F) {
        const int g = idx >> 6;
        ge[idx] = sums[idx] / fmaxf(cnts[g], 1.0f);
    }
}

// ---------------------------------------------------------------------------
// 7) graph_emb = tanh(ge @ W1 + b1) via WMMA f32.
//    Single 512-thread block = 16 waves = all 8x2 tiles.
// ---------------------------------------------------------------------------
__global__ void gemm2_wmma(const float* __restrict__ ge,
                           const float* __restrict__ W1,
                           const float* __restrict__ b1,
                           float* __restrict__ out) {
    const int wave = threadIdx.x >> 5;             // 0..15
    const int lane = threadIdx.x & 31;
    const int rt   = wave >> 1;                    // 0..7
    const int ct   = wave & 1;                     // 0..1
    const int half = lane >> 4;
    const int l15  = lane & 15;
    const int row  = rt * 16 + l15;
    const int col  = ct * 16 + l15;

    v8f acc = {};
    const float* __restrict__ grow = ge + (size_t)row * HID_F;

    #pragma unroll 4
    for (int k0 = 0; k0 < HID_F; k0 += 4) {
        const int k = k0 + 2 * half;
        v2f a = *(const v2f*)(grow + k);
        v2f b;
        b.x = W1[(size_t)k       * OUT_F + col];
        b.y = W1[(size_t)(k + 1) * OUT_F + col];
        acc = __builtin_amdgcn_wmma_f32_16x16x4_f32(
            false, a, false, b, (short)0, acc, false, false);
    }

    const float bias = b1[col];
    #pragma unroll
    for (int v = 0; v < 8; ++v) {
        const int m = rt * 16 + v + 8 * half;
        out[(size_t)m * OUT_F + col] = tanhf(acc[v] + bias);
    }
}

// ---------------------------------------------------------------------------
// Launch
// ---------------------------------------------------------------------------
extern "C" void kernel_launch(void* const* d_in, const int* in_sizes, int n_in,
                              void* d_out, int out_size, void* d_ws, size_t ws_size,
                              hipStream_t stream) {
    const float* x   = (const float*)d_in[0];                 // [N, 128]
    const int*   ei  = (const int*)d_in[1];                   // [2, E] flat
    const float* ew  = (const float*)d_in[2];                 // [E]
    const int*   bat = (const int*)d_in[3];                   // [N]
    // d_in[4] = num_graphs (known constant 128)
    const float* W   = (const float*)d_in[5];                 // [128, 64]
    const float* b   = (const float*)d_in[6];                 // [64]
    const float* W1  = (const float*)d_in[7];                 // [64, 32]
    const float* b1  = (const float*)d_in[8];                 // [32]

    const int* src = ei;
    const int* dst = ei + N_EDGES;

    // Output layout: [graph_emb 128*32 | node_emb 50000*64]
    float* graph_out = (float*)d_out;
    float* node_out  = (float*)d_out + (size_t)N_GRAPHS * OUT_F;  // agg buffer

    // Workspace layout (floats)
    float* h    = (float*)d_ws;                               // N*64
    float* deg  = h    + (size_t)N_NODES * HID_F;             // N
    float* dinv = deg  + N_NODES;                             // N
    float* sums = dinv + N_NODES;                             // G*64
    float* cnts = sums + (size_t)N_GRAPHS * HID_F;            // G
    float* ge   = cnts + N_GRAPHS;                            // G*64

    // Zero accumulators (sums & cnts contiguous -> one memset)
    hipMemsetAsync(deg, 0, (size_t)N_NODES * sizeof(float), stream);
    hipMemsetAsync(sums, 0, ((size_t)N_GRAPHS * HID_F + N_GRAPHS) * sizeof(float), stream);
    hipMemsetAsync(node_out, 0, (size_t)N_NODES * HID_F * sizeof(float), stream);

    // 1) degrees
    degree_kernel<<<(N_EDGES + 255) / 256, 256, 0, stream>>>(dst, ew, deg);
    // 2) dinv
    dinv_kernel<<<(N_NODES + 255) / 256, 256, 0, stream>>>(deg, dinv);
    // 3) h = x @ W (WMMA, 8 row-strips per block)
    gemm1_wmma<<<(ROW_TILES + 7) / 8, 256, 0, stream>>>(x, W, h);
    // 4) scatter messages (float4 per thread)
    {
        size_t total = (size_t)N_EDGES * (HID_F / 4);
        scatter_kernel<<<(unsigned)((total + 255) / 256), 256, 0, stream>>>(
            h, dinv, ew, src, dst, node_out);
    }
    // 5) finalize + pooling accumulation (float4 per thread)
    {
        size_t total = (size_t)N_NODES * (HID_F / 4);
        finalize_kernel<<<(unsigned)((total + 255) / 256), 256, 0, stream>>>(
            h, dinv, b, bat, node_out, sums, cnts);
    }
    // 6) mean pooling
    pool_mean_kernel<<<(N_GRAPHS * HID_F + 255) / 256, 256, 0, stream>>>(sums, cnts, ge);
    // 7) graph head (WMMA, one block of 16 waves)
    gemm2_wmma<<<1, 512, 0, stream>>>(ge, W1, b1, graph_out);
}